// ColumnNetworkXOR_46050639347942
// MI455X (gfx1250) — compile-verified
//
#include <hip/hip_runtime.h>
#include <math.h>

typedef float v2f __attribute__((ext_vector_type(2)));
typedef float v8f __attribute__((ext_vector_type(8)));

#define NPOP  24
#define WAVES 8
#define BLOCK (WAVES * 32)

__launch_bounds__(BLOCK)
__global__ void colnet_deriv_kernel(
    const float* __restrict__ t,
    const float* __restrict__ state,     // [B][72]
    const float* __restrict__ time_vec,  // [T]
    const float* __restrict__ stim,      // [T][2]
    const float* __restrict__ W0,        // [2][16]
    const float* __restrict__ W1,        // [2][8]
    const float* __restrict__ R0,        // [16][16]
    const float* __restrict__ R1,        // [8][8]
    const float* __restrict__ bg_w0,     // [16]
    const float* __restrict__ bg_w1,     // [8]
    const float* __restrict__ res,       // [24]
    const float* __restrict__ tau_m,     // [24]
    const float* __restrict__ adapt,     // [24]
    const float* __restrict__ tau_a,     // [24]
    const float* __restrict__ tau_s,     // [24]
    float* __restrict__ out,             // [B][72]
    int Btot, int T)
{
    __shared__ float s_fr[WAVES][32][26];    // fr tiles (stride 26: 8B-aligned rows, bank-spread)
    __shared__ float s_rec0[WAVES][32][16];  // WMMA D for rec0
    __shared__ float s_rec1[WAVES][32][16];  // WMMA D for rec1 (cols 8-15 unused pad)
    __shared__ float s_R0[16][16];           // R0 staged (B[k][n] = R0[n][k])
    __shared__ float s_R1p[16][8];           // R1 zero-padded to 16 rows
    // column-uniform coefficients (all divides hoisted here):
    __shared__ float s_P1[NPOP];  // res/tau_m            (ff multiplier)
    __shared__ float s_P2[NPOP];  // bg*8*tau_s*res/tau_m (constant term)
    __shared__ float s_P3[NPOP];  // tau_s*res/tau_m      (rec multiplier)
    __shared__ float s_P4[NPOP];  // 1/tau_m              (v multiplier)
    __shared__ float s_AR[NPOP];  // adapt/tau_a
    __shared__ float s_RA[NPOP];  // 1/tau_a
    __shared__ float s_RS[NPOP];  // 1/tau_s
    __shared__ float s_ext[2];

    const int tx   = threadIdx.x;
    const int wave = tx >> 5;
    const int lane = tx & 31;
    const int tl   = lane & 15;   // lane within half-wave
    const int hi   = lane >> 4;   // 0: lanes 0-15, 1: lanes 16-31

    // ---- stage small matrices + uniform coefficients into LDS ----
    for (int i = tx; i < 256; i += BLOCK) ((float*)s_R0)[i] = R0[i];
    for (int i = tx; i < 128; i += BLOCK) ((float*)s_R1p)[i] = (i < 64) ? R1[i] : 0.f;
    if (tx < NPOP) {
        float rtm = 1.f / tau_m[tx];
        float rs  = res[tx];
        float tss = tau_s[tx];
        float bg  = (tx < 16) ? bg_w0[tx] : bg_w1[tx - 16];
        s_P4[tx] = rtm;
        s_P1[tx] = rs * rtm;
        s_P3[tx] = tss * rs * rtm;
        s_P2[tx] = bg * 8.f * tss * rs * rtm;
        s_RA[tx] = 1.f / tau_a[tx];
        s_AR[tx] = adapt[tx] / tau_a[tx];
        s_RS[tx] = 1.f / tss;
    }

    // ---- grid-uniform stimulus interpolation (jnp.interp with edge clamp) ----
    if (tx == 0) {
        float ts = t[0];
        float e0, e1;
        if (ts <= time_vec[0])            { e0 = stim[0];           e1 = stim[1]; }
        else if (ts >= time_vec[T - 1])   { e0 = stim[2 * (T - 1)]; e1 = stim[2 * (T - 1) + 1]; }
        else {
            int i = 0;
            for (int j = 0; j < T - 1; ++j) if (time_vec[j + 1] <= ts) i = j + 1;
            if (i > T - 2) i = T - 2;
            float den = time_vec[i + 1] - time_vec[i];
            float f   = (den != 0.f) ? (ts - time_vec[i]) / den : 0.f;
            e0 = stim[2 * i]     + f * (stim[2 * (i + 1)]     - stim[2 * i]);
            e1 = stim[2 * i + 1] + f * (stim[2 * (i + 1) + 1] - stim[2 * i + 1]);
        }
        s_ext[0] = 20.f * e0;
        s_ext[1] = 20.f * e1;
    }

    __syncthreads();   // coefficients / matrices ready

    // ---- each lane owns one batch row ----
    long long row = (long long)blockIdx.x * BLOCK + tx;
    if (row >= Btot) row = Btot - 1;          // keep EXEC full for WMMA (dup write ok)
    const float4* s4 = (const float4*)(state + row * 72);
    float4*       o4 = (float4*)(out + row * 72);

    float v[NPOP], fr[NPOP];

    #pragma unroll
    for (int i = 0; i < 6; ++i) {
        float4 qv = s4[i];          // v[4i..4i+3]
        float4 qa = s4[6 + i];      // a
        float4 qp = s4[12 + i];     // prev_fr
        float4 qda, qdf;

        #define FRSTEP(C, KK)                                                  \
        {   const int kk = (KK);                                               \
            float x = qv.C - qa.C;                                             \
            float d = -expm1f(-x);                                             \
            float f = (fabsf(x) < 1e-6f) ? (1.0f + 0.5f * x) : (x / d);        \
            v[kk]  = qv.C;                                                     \
            fr[kk] = f;                                                        \
            qda.C = s_AR[kk] * f - qa.C * s_RA[kk];                            \
            qdf.C = (f - qp.C) * s_RS[kk];                                     \
        }
        FRSTEP(x, 4 * i + 0)
        FRSTEP(y, 4 * i + 1)
        FRSTEP(z, 4 * i + 2)
        FRSTEP(w, 4 * i + 3)
        #undef FRSTEP

        // fr -> LDS as two contiguous 8B stores (ds_store_b64)
        v2f p0, p1;
        p0.x = fr[4 * i + 0]; p0.y = fr[4 * i + 1];
        p1.x = fr[4 * i + 2]; p1.y = fr[4 * i + 3];
        *(v2f*)(&s_fr[wave][lane][4 * i + 0]) = p0;
        *(v2f*)(&s_fr[wave][lane][4 * i + 2]) = p1;

        o4[6 + i]  = qda;   // da  -> cols 24..47
        o4[12 + i] = qdf;   // dfr -> cols 48..71
    }

    __syncthreads();   // fr tiles ready

    // ---- B fragments: tile-invariant, contiguous v2f (ds_load_b64) ----
    // A 16x4 f32 layout (ISA 7.12.2): lanes 0-15 M=lane {VGPR0:K=0, VGPR1:K=1},
    //                                 lanes 16-31       {VGPR0:K=2, VGPR1:K=3}.
    // B 4x16: B[k][n] = R[n][k]; same VGPR/lane striping with N=tl.
    const int kk0 = 2 * hi;    // fragment base: k0 = 4s + 2*hi, k1 = k0 + 1
    v2f b0[4], b1[2];
    #pragma unroll
    for (int s = 0; s < 4; ++s) b0[s] = *(const v2f*)(&s_R0[tl][4 * s + kk0]);
    #pragma unroll
    for (int s = 0; s < 2; ++s) b1[s] = *(const v2f*)(&s_R1p[tl][4 * s + kk0]);

    // ---- WMMA: per wave, two 16-row tiles; D = A(fr) x B(R^T) in pure f32 ----
    #pragma unroll
    for (int tile = 0; tile < 2; ++tile) {
        const float* frW = &s_fr[wave][tile * 16 + tl][0];

        v8f c0 = {0.f, 0.f, 0.f, 0.f, 0.f, 0.f, 0.f, 0.f};
        #pragma unroll
        for (int s = 0; s < 4; ++s) {            // K = 16 for rec0
            v2f aV = *(const v2f*)(frW + 4 * s + kk0);
            c0 = __builtin_amdgcn_wmma_f32_16x16x4_f32(
                     false, aV, false, b0[s], (short)0, c0, false, false);
        }

        v8f c1 = {0.f, 0.f, 0.f, 0.f, 0.f, 0.f, 0.f, 0.f};
        #pragma unroll
        for (int s = 0; s < 2; ++s) {            // K = 8 for rec1 (N>=8 cols zero)
            v2f aV = *(const v2f*)(frW + 16 + 4 * s + kk0);
            c1 = __builtin_amdgcn_wmma_f32_16x16x4_f32(
                     false, aV, false, b1[s], (short)0, c1, false, false);
        }

        // D layout: VGPR r -> (M = r + 8*hi, N = tl); unconditional stores
        #pragma unroll
        for (int r = 0; r < 8; ++r) {
            s_rec0[wave][tile * 16 + r + hi * 8][tl] = c0[r];
            s_rec1[wave][tile * 16 + r + hi * 8][tl] = c1[r];
        }
    }

    __syncthreads();

    // ---- finish dv for own row (all divides folded into s_P* coefficients) ----
    const float ext0 = s_ext[0], ext1 = s_ext[1];
    const float4* r0p = (const float4*)&s_rec0[wave][lane][0];  // ds_load_b128 x4
    const float4* r1p = (const float4*)&s_rec1[wave][lane][0];  // ds_load_b128 x2
    float rec[NPOP];
    #pragma unroll
    for (int i = 0; i < 4; ++i) {
        float4 q = r0p[i];
        rec[4 * i + 0] = q.x; rec[4 * i + 1] = q.y;
        rec[4 * i + 2] = q.z; rec[4 * i + 3] = q.w;
    }
    #pragma unroll
    for (int i = 0; i < 2; ++i) {
        float4 q = r1p[i];
        rec[16 + 4 * i + 0] = q.x; rec[16 + 4 * i + 1] = q.y;
        rec[16 + 4 * i + 2] = q.z; rec[16 + 4 * i + 3] = q.w;
    }

    float dv[NPOP];
    #pragma unroll
    for (int n = 0; n < 16; ++n) {
        float ff0 = ext0 * W0[n] + ext1 * W0[16 + n];
        dv[n] = s_P2[n] + ff0 * s_P1[n] + rec[n] * s_P3[n] - v[n] * s_P4[n];
    }
    const float f10_0 = 10.f * fr[0], f10_8 = 10.f * fr[8];
    #pragma unroll
    for (int n = 0; n < 8; ++n) {
        float ff1 = f10_0 * W1[n] + f10_8 * W1[8 + n];
        const int m = 16 + n;
        dv[m] = s_P2[m] + ff1 * s_P1[m] + rec[m] * s_P3[m] - v[m] * s_P4[m];
    }

    #pragma unroll
    for (int i = 0; i < 6; ++i) {
        float4 q;
        q.x = dv[4 * i + 0]; q.y = dv[4 * i + 1];
        q.z = dv[4 * i + 2]; q.w = dv[4 * i + 3];
        o4[i] = q;           // dv -> cols 0..23
    }
}

extern "C" void kernel_launch(void* const* d_in, const int* in_sizes, int n_in,
                              void* d_out, int out_size, void* d_ws, size_t ws_size,
                              hipStream_t stream) {
    const float* t        = (const float*)d_in[0];
    const float* state    = (const float*)d_in[1];
    const float* time_vec = (const float*)d_in[2];
    const float* stim     = (const float*)d_in[3];
    const float* W0       = (const float*)d_in[4];
    const float* W1       = (const float*)d_in[5];
    const float* R0       = (const float*)d_in[6];
    const float* R1       = (const float*)d_in[7];
    const float* bg_w0    = (const float*)d_in[8];
    const float* bg_w1    = (const float*)d_in[9];
    const float* res      = (const float*)d_in[10];
    const float* tau_m    = (const float*)d_in[11];
    const float* adapt    = (const float*)d_in[12];
    const float* tau_a    = (const float*)d_in[13];
    const float* tau_s    = (const float*)d_in[14];
    float* out = (float*)d_out;

    const int Btot   = in_sizes[1] / 72;
    const int T      = in_sizes[2];
    const int blocks = (Btot + BLOCK - 1) / BLOCK;

    colnet_deriv_kernel<<<blocks, BLOCK, 0, stream>>>(
        t, state, time_vec, stim, W0, W1, R0, R1, bg_w0, bg_w1,
        res, tau_m, adapt, tau_a, tau_s, out, Btot, T);
}